// CRF_38379827757404
// MI455X (gfx1250) — compile-verified
//
#include <hip/hip_runtime.h>
#include <math.h>

// ---------------------------------------------------------------------------
// Fused mean-field CRF for MI455X (gfx1250, wave32).
//
// Never materializes the two 8192x8192 Gaussian kernels (512 MB). Each
// 16-row output tile is owned by one 256-thread workgroup (8 waves); the 512
// 16-column tiles are split across the waves (64 each). Per tile, the
// feature Gram matrix is computed with V_WMMA_F32_16X16X4_F32 (two chained
// WMMAs for the 5-d bilateral features, one for the 3-d spatial features),
// expanded to squared distances, passed through v_exp_f32, and the 4-channel
// filter responses accumulate in registers. Wave partials combine through a
// small LDS reduction. All data is stored point-interleaved (fb[n][8],
// fs[n][4], sq[n][2], qs[n][8], sc[n][2]) so per-tile operand fetches are
// b64/b128 loads. Working set (~800 KB) is L2-resident; the pass is
// VALU/TRANS bound with 4096 waves in flight for latency hiding.
// ---------------------------------------------------------------------------

typedef float v2f __attribute__((ext_vector_type(2)));
typedef float v4f __attribute__((ext_vector_type(4)));
typedef float v8f __attribute__((ext_vector_type(8)));

#define CRF_N      8192
#define CRF_C      4
#define CRF_ALPHA  5.0f
#define CRF_BETA   5.0f
#define CRF_GAMMA  5.0f
#define CRF_W1     1.0f
#define CRF_W2     1.0f
#define CRF_ITER   5
#define NTILES     (CRF_N / 16)    /* 512 */
#define WPB        8               /* waves per block */
#define TPW        (NTILES / WPB)  /* 64 m-tiles per wave */

static __device__ __forceinline__ v8f wmma4(v2f a, v2f b, v8f c) {
  // D(16x16,f32) = A(16x4,f32) x B(4x16,f32) + C
  return __builtin_amdgcn_wmma_f32_16x16x4_f32(
      /*neg_a=*/false, a, /*neg_b=*/false, b,
      /*c_mod=*/(short)0, c, /*reuse_a=*/false, /*reuse_b=*/false);
}

// ---------------------------------------------------------------------------
// Pass 0: point-interleaved padded features, squared norms, q0 = softmax(lu).
// fb[n][8]: 3 spatial/ALPHA, 2 img/BETA, 3 zero pad   (bilateral)
// fs[n][4]: 3 spatial/GAMMA, 1 zero pad               (spatial)
// sqp[n][2]: |f1|^2, |f2|^2
// ---------------------------------------------------------------------------
__global__ void crf_prep(const float* __restrict__ lu,
                         const float* __restrict__ feat,
                         float* __restrict__ fb, float* __restrict__ fs,
                         float* __restrict__ sqp, float* __restrict__ q) {
  int n = blockIdx.x * blockDim.x + threadIdx.x;
  if (n >= CRF_N) return;
  float x = (float)(n >> 8);
  float y = (float)((n >> 4) & 15);
  float z = (float)(n & 15);

  float a0 = x * (1.0f / CRF_ALPHA);
  float a1 = y * (1.0f / CRF_ALPHA);
  float a2 = z * (1.0f / CRF_ALPHA);
  float b0 = feat[0 * CRF_N + n] * (1.0f / CRF_BETA);
  float b1 = feat[1 * CRF_N + n] * (1.0f / CRF_BETA);
  v4f* fb4 = (v4f*)fb;
  fb4[2 * n + 0] = (v4f){a0, a1, a2, b0};
  fb4[2 * n + 1] = (v4f){b1, 0.0f, 0.0f, 0.0f};

  float g0 = x * (1.0f / CRF_GAMMA);
  float g1 = y * (1.0f / CRF_GAMMA);
  float g2 = z * (1.0f / CRF_GAMMA);
  ((v4f*)fs)[n] = (v4f){g0, g1, g2, 0.0f};

  float s1 = a0 * a0 + a1 * a1 + a2 * a2 + b0 * b0 + b1 * b1;
  float s2 = g0 * g0 + g1 * g1 + g2 * g2;
  ((v2f*)sqp)[n] = (v2f){s1, s2};

  float l0 = lu[0 * CRF_N + n], l1 = lu[1 * CRF_N + n];
  float l2 = lu[2 * CRF_N + n], l3 = lu[3 * CRF_N + n];
  float mx = fmaxf(fmaxf(l0, l1), fmaxf(l2, l3));
  float e0 = __expf(l0 - mx), e1 = __expf(l1 - mx);
  float e2 = __expf(l2 - mx), e3 = __expf(l3 - mx);
  float inv = 1.0f / (e0 + e1 + e2 + e3);
  q[0 * CRF_N + n] = e0 * inv;
  q[1 * CRF_N + n] = e1 * inv;
  q[2 * CRF_N + n] = e2 * inv;
  q[3 * CRF_N + n] = e3 * inv;
}

// ---------------------------------------------------------------------------
// Pass 1: scale = rsqrt(row sums of K1/K2). One block per i-tile, m-tiles
// split over 8 waves, LDS combine.
// ---------------------------------------------------------------------------
__global__ void __launch_bounds__(256) crf_norm(
    const float* __restrict__ fb, const float* __restrict__ fs,
    const float* __restrict__ sqp, float* __restrict__ scp) {
  __shared__ float red[WPB * 2 * 16];  // [wave][kernel][row]

  const int lane = threadIdx.x & 31;
  const int wave = threadIdx.x >> 5;
  const int i0 = blockIdx.x << 4;
  const int hl = lane >> 4;
  const int col = lane & 15;

  const v2f* fb2 = (const v2f*)fb;   // fb2[n*4 + d2]  (d2 = feature pair)
  const v2f* fs2 = (const v2f*)fs;   // fs2[n*2 + d2]
  const v2f* sq2 = (const v2f*)sqp;  // sq2[n] = {sq1, sq2}

  const int ia = i0 + col;
  v2f a1lo = fb2[ia * 4 + hl];
  v2f a1hi = fb2[ia * 4 + 2 + hl];
  v2f a2   = fs2[ia * 2 + hl];

  v2f sqi[8];
#pragma unroll
  for (int r = 0; r < 8; ++r) sqi[r] = sq2[i0 + 8 * hl + r];

  const v8f vzero = {0, 0, 0, 0, 0, 0, 0, 0};
  v8f rs1 = vzero, rs2 = vzero;

  for (int mt = wave * TPW; mt < wave * TPW + TPW; ++mt) {
    const int j = (mt << 4) + col;
    v2f b1lo = fb2[j * 4 + hl];
    v2f b1hi = fb2[j * 4 + 2 + hl];
    v2f b2   = fs2[j * 2 + hl];

    v8f g1 = wmma4(a1lo, b1lo, vzero);
    g1 = wmma4(a1hi, b1hi, g1);          // 5-d bilateral Gram (padded to 8)
    v8f g2 = wmma4(a2, b2, vzero);       // 3-d spatial Gram (padded to 4)

    v2f sqj = sq2[j];
#pragma unroll
    for (int r = 0; r < 8; ++r) {
      float d1 = fmaxf(sqi[r].x + sqj.x - 2.0f * g1[r], 0.0f);
      float d2 = fmaxf(sqi[r].y + sqj.y - 2.0f * g2[r], 0.0f);
      rs1[r] += __expf(-0.5f * d1);
      rs2[r] += __expf(-0.5f * d2);
    }
  }

#pragma unroll
  for (int m = 1; m <= 8; m <<= 1) {
#pragma unroll
    for (int r = 0; r < 8; ++r) {
      rs1[r] += __shfl_xor(rs1[r], m, 32);
      rs2[r] += __shfl_xor(rs2[r], m, 32);
    }
  }
  if (col == 0) {
#pragma unroll
    for (int r = 0; r < 8; ++r) {
      red[(wave * 2 + 0) * 16 + 8 * hl + r] = rs1[r];
      red[(wave * 2 + 1) * 16 + 8 * hl + r] = rs2[r];
    }
  }
  __syncthreads();
  if (threadIdx.x < 16) {
    const int t = threadIdx.x;
    float s1 = 0.0f, s2 = 0.0f;
#pragma unroll
    for (int w = 0; w < WPB; ++w) {
      s1 += red[(w * 2 + 0) * 16 + t];
      s2 += red[(w * 2 + 1) * 16 + t];
    }
    ((v2f*)scp)[i0 + t] = (v2f){rsqrtf(s1), rsqrtf(s2)};
  }
}

// ---------------------------------------------------------------------------
// Stage qs[n][8] = {q*scale1 (4ch), q*scale2 (4ch)} so crf_iter can
// overwrite q in place and fetch all 8 values with two b128 loads.
// ---------------------------------------------------------------------------
__global__ void crf_qscale(const float* __restrict__ q,
                           const float* __restrict__ scp,
                           float* __restrict__ qsp) {
  int n = blockIdx.x * blockDim.x + threadIdx.x;
  if (n >= CRF_N) return;
  v2f s = ((const v2f*)scp)[n];
  v4f qv = {q[0 * CRF_N + n], q[1 * CRF_N + n],
            q[2 * CRF_N + n], q[3 * CRF_N + n]};
  v4f* qsp4 = (v4f*)qsp;
  qsp4[2 * n + 0] = qv * s.x;
  qsp4[2 * n + 1] = qv * s.y;
}

// ---------------------------------------------------------------------------
// One mean-field iteration, fully fused per 16-row tile:
// kernel recompute (WMMA Gram + exp) -> filter -> compat -> softmax -> q.
// ---------------------------------------------------------------------------
__global__ void __launch_bounds__(256) crf_iter(
    const float* __restrict__ fb, const float* __restrict__ fs,
    const float* __restrict__ sqp, const float* __restrict__ scp,
    const float* __restrict__ qsp, const float* __restrict__ lu,
    const float* __restrict__ comp, float* __restrict__ q) {
  __shared__ float red[WPB * 2 * CRF_C * 16];  // [wave][kernel][c][row] 4KB

  const int lane = threadIdx.x & 31;
  const int wave = threadIdx.x >> 5;
  const int i0 = blockIdx.x << 4;
  const int hl = lane >> 4;
  const int col = lane & 15;

  const v2f* fb2 = (const v2f*)fb;
  const v2f* fs2 = (const v2f*)fs;
  const v2f* sq2 = (const v2f*)sqp;
  const v4f* qs4 = (const v4f*)qsp;

  const int ia = i0 + col;
  v2f a1lo = fb2[ia * 4 + hl];
  v2f a1hi = fb2[ia * 4 + 2 + hl];
  v2f a2   = fs2[ia * 2 + hl];

  v2f sqi[8];
#pragma unroll
  for (int r = 0; r < 8; ++r) sqi[r] = sq2[i0 + 8 * hl + r];

  const v8f vzero = {0, 0, 0, 0, 0, 0, 0, 0};
  v8f acc1[CRF_C] = {vzero, vzero, vzero, vzero};
  v8f acc2[CRF_C] = {vzero, vzero, vzero, vzero};

  for (int mt = wave * TPW; mt < wave * TPW + TPW; ++mt) {
    const int j = (mt << 4) + col;
    v2f b1lo = fb2[j * 4 + hl];
    v2f b1hi = fb2[j * 4 + 2 + hl];
    v2f b2   = fs2[j * 2 + hl];

    v8f g1 = wmma4(a1lo, b1lo, vzero);
    g1 = wmma4(a1hi, b1hi, g1);
    v8f g2 = wmma4(a2, b2, vzero);

    v2f sqj = sq2[j];
    v4f w1 = qs4[j * 2 + 0];   // q*scale1, lane's fixed column j
    v4f w2 = qs4[j * 2 + 1];   // q*scale2

    v8f kv1, kv2;
#pragma unroll
    for (int r = 0; r < 8; ++r) {
      float d1 = fmaxf(sqi[r].x + sqj.x - 2.0f * g1[r], 0.0f);
      float d2 = fmaxf(sqi[r].y + sqj.y - 2.0f * g2[r], 0.0f);
      kv1[r] = __expf(-0.5f * d1);
      kv2[r] = __expf(-0.5f * d2);
    }
#pragma unroll
    for (int c = 0; c < CRF_C; ++c) {
      acc1[c] += kv1 * w1[c];
      acc2[c] += kv2 * w2[c];
    }
  }

  // intra-wave: combine the 16 lane-columns of each half
#pragma unroll
  for (int m = 1; m <= 8; m <<= 1) {
#pragma unroll
    for (int c = 0; c < CRF_C; ++c) {
#pragma unroll
      for (int r = 0; r < 8; ++r) {
        acc1[c][r] += __shfl_xor(acc1[c][r], m, 32);
        acc2[c][r] += __shfl_xor(acc2[c][r], m, 32);
      }
    }
  }

  if (col == 0) {
#pragma unroll
    for (int c = 0; c < CRF_C; ++c) {
#pragma unroll
      for (int r = 0; r < 8; ++r) {
        red[((wave * 2 + 0) * CRF_C + c) * 16 + 8 * hl + r] = acc1[c][r];
        red[((wave * 2 + 1) * CRF_C + c) * 16 + 8 * hl + r] = acc2[c][r];
      }
    }
  }
  __syncthreads();

  if (threadIdx.x < 16) {
    const int t = threadIdx.x;
    const int i = i0 + t;
    v2f sc = ((const v2f*)scp)[i];
    float qc[CRF_C];
#pragma unroll
    for (int c = 0; c < CRF_C; ++c) {
      float s1 = 0.0f, s2 = 0.0f;
#pragma unroll
      for (int w = 0; w < WPB; ++w) {
        s1 += red[((w * 2 + 0) * CRF_C + c) * 16 + t];
        s2 += red[((w * 2 + 1) * CRF_C + c) * 16 + t];
      }
      qc[c] = CRF_W1 * s1 * sc.x + CRF_W2 * s2 * sc.y;
    }
    float lg[CRF_C];
#pragma unroll
    for (int o = 0; o < CRF_C; ++o) {
      float u = 0.0f;
#pragma unroll
      for (int c = 0; c < CRF_C; ++c) u += comp[o * 4 + c] * qc[c];
      lg[o] = lu[o * CRF_N + i] - u;
    }
    float mx = fmaxf(fmaxf(lg[0], lg[1]), fmaxf(lg[2], lg[3]));
    float e[CRF_C];
    float s = 0.0f;
#pragma unroll
    for (int o = 0; o < CRF_C; ++o) { e[o] = __expf(lg[o] - mx); s += e[o]; }
    float inv = 1.0f / s;
#pragma unroll
    for (int o = 0; o < CRF_C; ++o) q[o * CRF_N + i] = e[o] * inv;
  }
}

// ---------------------------------------------------------------------------
extern "C" void kernel_launch(void* const* d_in, const int* in_sizes, int n_in,
                              void* d_out, int out_size, void* d_ws, size_t ws_size,
                              hipStream_t stream) {
  (void)in_sizes; (void)n_in; (void)out_size; (void)ws_size;
  const float* lu   = (const float*)d_in[0];  // [4][8192]
  const float* feat = (const float*)d_in[1];  // [2][8192]
  const float* comp = (const float*)d_in[2];  // [4][4]
  float* q  = (float*)d_out;                  // [4][8192], lives here throughout
  float* ws = (float*)d_ws;                   // 24*N floats = 768 KB

  float* fb  = ws;                 // 8*N  point-interleaved bilateral features
  float* fs  = fb + 8 * CRF_N;     // 4*N  point-interleaved spatial features
  float* sqp = fs + 4 * CRF_N;     // 2*N  packed |f|^2 pairs
  float* scp = sqp + 2 * CRF_N;    // 2*N  packed rsqrt-norm pairs
  float* qsp = scp + 2 * CRF_N;    // 8*N  staged q*scale (both kernels)

  crf_prep<<<CRF_N / 256, 256, 0, stream>>>(lu, feat, fb, fs, sqp, q);
  crf_norm<<<NTILES, 256, 0, stream>>>(fb, fs, sqp, scp);
  for (int it = 0; it < CRF_ITER; ++it) {
    crf_qscale<<<CRF_N / 256, 256, 0, stream>>>(q, scp, qsp);
    crf_iter<<<NTILES, 256, 0, stream>>>(fb, fs, sqp, scp, qsp, lu, comp, q);
  }
}